// LSTMDecoderV2_88313117540521
// MI455X (gfx1250) — compile-verified
//
#include <hip/hip_runtime.h>
#include <hip/hip_bf16.h>
#include <math.h>

// Problem constants (from reference)
#define B_   32
#define T_   128
#define V_   16000
#define E_   512
#define H_   1024
#define F_   2048
#define G4H  4096     // 4*H
#define BT   4096     // B*T
#define START_IDX_ 1

typedef float v2f __attribute__((ext_vector_type(2)));
typedef float v8f __attribute__((ext_vector_type(8)));

// D = A(16x4) * B(4x16) + C, fp32 WMMA (CDNA5). 8-arg form:
// (neg_a, A, neg_b, B, c_mod, C, reuse_a, reuse_b)
__device__ __forceinline__ v8f wmma4(v2f a, v2f b, v8f c) {
  return __builtin_amdgcn_wmma_f32_16x16x4_f32(false, a, false, b, (short)0, c,
                                               false, false);
}

// Two chained WMMAs fed from one float4 per operand. K-order is permuted
// (lane-half holds K {k..k+3} / {k+4..k+7}) identically for A and B, which is
// valid because GEMM is a reduction over K.
__device__ __forceinline__ v8f wmma8(float4 a, float4 b, v8f c) {
  v2f a01 = {a.x, a.y}, a23 = {a.z, a.w};
  v2f b01 = {b.x, b.y}, b23 = {b.z, b.w};
  c = wmma4(a01, b01, c);
  c = wmma4(a23, b23, c);
  return c;
}

__device__ __forceinline__ float sigmoidf_(float x) {
  return 1.0f / (1.0f + __expf(-x));
}

// ---------------------------------------------------------------------------
// 1) pooled[b][f] = mean over 7x7 spatial
// ---------------------------------------------------------------------------
__global__ void pool_kernel(const float* __restrict__ feat,
                            float* __restrict__ pooled) {
  int i = blockIdx.x * blockDim.x + threadIdx.x;
  if (i >= B_ * F_) return;
  const float* p = feat + (size_t)i * 49;
  float s = 0.f;
#pragma unroll
  for (int j = 0; j < 49; ++j) s += p[j];
  pooled[i] = s * (1.0f / 49.0f);
}

// ---------------------------------------------------------------------------
// 2) init = pooled @ fc_W.T + fc_b ; split into h0 / c0
//    M=32 (2 tiles) x N=2048 (128 tiles), K=2048. One wave per 16x16 tile.
// ---------------------------------------------------------------------------
__global__ void init_gemm_kernel(const float* __restrict__ pooled,
                                 const float* __restrict__ fcW,
                                 const float* __restrict__ fcb,
                                 float* __restrict__ hcur,
                                 float* __restrict__ ccur) {
  int wid  = blockIdx.x * 8 + (threadIdx.x >> 5);   // 256 waves total
  int lane = threadIdx.x & 31;
  int lm = lane & 15, lh = lane >> 4;
  int m0 = (wid >> 7) * 16;          // 2 M-tiles
  int n0 = (wid & 127) * 16;         // 128 N-tiles
  const float4* arow = (const float4*)(pooled + (size_t)(m0 + lm) * F_) + lh;
  const float4* brow = (const float4*)(fcW    + (size_t)(n0 + lm) * F_) + lh;
  v8f acc = {};
#pragma unroll 4
  for (int k8 = 0; k8 < F_ / 8; ++k8) {
    acc = wmma8(arow[2 * k8], brow[2 * k8], acc);
  }
  int col = n0 + lm;
  float bias = fcb[col];
#pragma unroll
  for (int v = 0; v < 8; ++v) {
    int row = m0 + v + 8 * lh;       // batch index
    float val = acc[v] + bias;
    if (col < H_) hcur[row * H_ + col]        = val;
    else          ccur[row * H_ + (col - H_)] = val;
  }
}

// ---------------------------------------------------------------------------
// 3) Xg[b*T+t][:] = emb[in_idx(b,t)] @ W_ih.T + (b_ih + b_hh)
//    Gathered-A GEMM: M=4096 (256 tiles) x N=4096 (256 tiles), K=512.
// ---------------------------------------------------------------------------
__global__ void embed_gemm_kernel(const int* __restrict__ reports,
                                  const float* __restrict__ emb,
                                  const float* __restrict__ Wih,
                                  const float* __restrict__ bih,
                                  const float* __restrict__ bhh,
                                  float* __restrict__ Xg) {
  int wid  = blockIdx.x * 8 + (threadIdx.x >> 5);   // 65536 waves total
  int lane = threadIdx.x & 31;
  int lm = lane & 15, lh = lane >> 4;
  int m0 = (wid >> 8) * 16;
  int n0 = (wid & 255) * 16;
  int m = m0 + lm;
  int b = m >> 7;                    // m / T
  int t = m & (T_ - 1);
  int idx = (t == 0) ? START_IDX_ : reports[b * T_ + (t - 1)];
  const float4* arow = (const float4*)(emb + (size_t)idx * E_) + lh;
  const float4* brow = (const float4*)(Wih + (size_t)(n0 + lm) * E_) + lh;
  v8f acc = {};
#pragma unroll 4
  for (int k8 = 0; k8 < E_ / 8; ++k8) {
    acc = wmma8(arow[2 * k8], brow[2 * k8], acc);
  }
  int col = n0 + lm;
  float bias = bih[col] + bhh[col];
#pragma unroll
  for (int v = 0; v < 8; ++v) {
    int row = m0 + v + 8 * lh;
    Xg[(size_t)row * G4H + col] = acc[v] + bias;
  }
}

// ---------------------------------------------------------------------------
// 4a) gates = Xg[:,t,:] + h @ W_hh.T   (acc pre-loaded with Xg as C matrix)
//     M=32 (2 tiles) x N=4096 (256 tiles), K=1024. One wave per tile.
// ---------------------------------------------------------------------------
__global__ void gates_gemm_kernel(const float* __restrict__ hcur,
                                  const float* __restrict__ Whh,
                                  const float* __restrict__ Xg,
                                  float* __restrict__ gates, int t) {
  int wid  = blockIdx.x * 8 + (threadIdx.x >> 5);   // 512 waves total
  int lane = threadIdx.x & 31;
  int lm = lane & 15, lh = lane >> 4;
  int m0 = (wid >> 8) * 16;
  int n0 = (wid & 255) * 16;
  int col = n0 + lm;
  v8f acc;
#pragma unroll
  for (int v = 0; v < 8; ++v) {
    int row = m0 + v + 8 * lh;       // batch index
    acc[v] = Xg[(size_t)(row * T_ + t) * G4H + col];
  }
  const float4* arow = (const float4*)(hcur + (size_t)(m0 + lm) * H_) + lh;
  const float4* brow = (const float4*)(Whh  + (size_t)(n0 + lm) * H_) + lh;
#pragma unroll 4
  for (int k8 = 0; k8 < H_ / 8; ++k8) {
    acc = wmma8(arow[2 * k8], brow[2 * k8], acc);
  }
#pragma unroll
  for (int v = 0; v < 8; ++v) {
    int row = m0 + v + 8 * lh;
    gates[row * G4H + col] = acc[v];
  }
}

// ---------------------------------------------------------------------------
// 4b) LSTM pointwise cell update; writes h into hs[(b*T+t)].
// ---------------------------------------------------------------------------
__global__ void lstm_update_kernel(const float* __restrict__ gates,
                                   float* __restrict__ hcur,
                                   float* __restrict__ ccur,
                                   float* __restrict__ hs, int t) {
  int i = blockIdx.x * blockDim.x + threadIdx.x;
  if (i >= B_ * H_) return;
  int b = i >> 10;                   // i / H
  int j = i & (H_ - 1);
  const float* g = gates + (size_t)b * G4H;
  float ig = g[j];
  float fg = g[H_ + j];
  float gg = g[2 * H_ + j];
  float og = g[3 * H_ + j];
  float c = sigmoidf_(fg) * ccur[i] + sigmoidf_(ig) * tanhf(gg);
  float h = sigmoidf_(og) * tanhf(c);
  ccur[i] = c;
  hcur[i] = h;
  hs[(size_t)(b * T_ + t) * H_ + j] = h;
}

// ---------------------------------------------------------------------------
// 5) seq_out = hs @ Wv.T + bv
//    M=4096 (256 tiles) x N=16000. Each wave: 16x64 tile (A-frag reuse x4).
//    Output is streamed with non-temporal stores (write-once, 262 MB) so the
//    L2 keeps Wv (65 MB) and hs (17 MB) resident across the M-tile reuse.
// ---------------------------------------------------------------------------
__global__ void logits_gemm_kernel(const float* __restrict__ hs,
                                   const float* __restrict__ Wv,
                                   const float* __restrict__ bv,
                                   float* __restrict__ out) {
  int wid  = blockIdx.x * 8 + (threadIdx.x >> 5);   // 64000 waves total
  int lane = threadIdx.x & 31;
  int lm = lane & 15, lh = lane >> 4;
  int mt = wid / 250;                // 256 M-tiles
  int nb = wid % 250;                // 250 blocks of 64 columns
  int m0 = mt * 16;
  int n0 = nb * 64;
  const float4* arow = (const float4*)(hs + (size_t)(m0 + lm) * H_) + lh;
  const float4* b0 = (const float4*)(Wv + (size_t)(n0 + lm) * H_) + lh;
  const float4* b1 = b0 + (size_t)16 * (H_ / 4);
  const float4* b2 = b0 + (size_t)32 * (H_ / 4);
  const float4* b3 = b0 + (size_t)48 * (H_ / 4);
  v8f acc0 = {}, acc1 = {}, acc2 = {}, acc3 = {};
#pragma unroll 2
  for (int k8 = 0; k8 < H_ / 8; ++k8) {
    float4 a = arow[2 * k8];
    acc0 = wmma8(a, b0[2 * k8], acc0);
    acc1 = wmma8(a, b1[2 * k8], acc1);
    acc2 = wmma8(a, b2[2 * k8], acc2);
    acc3 = wmma8(a, b3[2 * k8], acc3);
  }
  v8f accs[4] = {acc0, acc1, acc2, acc3};
#pragma unroll
  for (int q = 0; q < 4; ++q) {
    int col = n0 + q * 16 + lm;
    float bias = bv[col];
    float* ocol = out + (size_t)m0 * V_ + col;
#pragma unroll
    for (int v = 0; v < 8; ++v) {
      __builtin_nontemporal_store(accs[q][v] + bias,
                                  ocol + (size_t)(v + 8 * lh) * V_);
    }
  }
}

// ---------------------------------------------------------------------------
extern "C" void kernel_launch(void* const* d_in, const int* in_sizes, int n_in,
                              void* d_out, int out_size, void* d_ws,
                              size_t ws_size, hipStream_t stream) {
  (void)in_sizes; (void)n_in; (void)out_size; (void)ws_size;

  const float* feat    = (const float*)d_in[0];   // (B,F,7,7)
  const int*   reports = (const int*)  d_in[1];   // (B,T)
  const float* fcW     = (const float*)d_in[2];   // (2H,F)
  const float* fcb     = (const float*)d_in[3];   // (2H,)
  const float* emb     = (const float*)d_in[4];   // (V,E)
  const float* Wih     = (const float*)d_in[5];   // (4H,E)
  const float* Whh     = (const float*)d_in[6];   // (4H,H)
  const float* bih     = (const float*)d_in[7];   // (4H,)
  const float* bhh     = (const float*)d_in[8];   // (4H,)
  const float* Wv      = (const float*)d_in[9];   // (V,H)
  const float* bv      = (const float*)d_in[10];  // (V,)
  float*       out     = (float*)d_out;           // (B,T,V)

  // Workspace carve-up (~85 MB of fp32)
  float* ws     = (float*)d_ws;
  float* pooled = ws;                       // B*F      = 65536
  float* hcur   = pooled + B_ * F_;         // B*H      = 32768
  float* ccur   = hcur + B_ * H_;           // B*H      = 32768
  float* gates  = ccur + B_ * H_;           // B*4H     = 131072
  float* hs     = gates + (size_t)B_ * G4H; // BT*H     = 4194304
  float* Xg     = hs + (size_t)BT * H_;     // BT*4H    = 16777216

  // 1) spatial mean pool
  pool_kernel<<<(B_ * F_ + 255) / 256, 256, 0, stream>>>(feat, pooled);

  // 2) init GEMM -> h0, c0   (256 waves)
  init_gemm_kernel<<<32, 256, 0, stream>>>(pooled, fcW, fcb, hcur, ccur);

  // 3) hoisted input-projection GEMM for all timesteps   (65536 waves)
  embed_gemm_kernel<<<8192, 256, 0, stream>>>(reports, emb, Wih, bih, bhh, Xg);

  // 4) sequential LSTM scan: 128 x (gates GEMM + pointwise update)
  for (int t = 0; t < T_; ++t) {
    gates_gemm_kernel<<<64, 256, 0, stream>>>(hcur, Whh, Xg, gates, t);
    lstm_update_kernel<<<(B_ * H_ + 255) / 256, 256, 0, stream>>>(gates, hcur,
                                                                  ccur, hs, t);
  }

  // 5) vocab projection GEMM (dominant: 134 GFLOP)   (64000 waves)
  logits_gemm_kernel<<<8000, 256, 0, stream>>>(hs, Wv, bv, out);
}